// EncoderModule_57732950393070
// MI455X (gfx1250) — compile-verified
//
#include <hip/hip_runtime.h>

// ---------------------------------------------------------------------------
// VQ-VAE audio encoder for MI455X (gfx1250, wave32).
//   conv stack: fp32 VALU direct convs (tiny channel counts, K=7)
//   VQ:         V_WMMA_F32_16X16X4_F32 GEMM (z . codebook) + fused argmin,
//               codebook chunks staged in LDS and shared by 16 waves/block.
// ---------------------------------------------------------------------------

typedef __attribute__((ext_vector_type(2))) float v2f;
typedef __attribute__((ext_vector_type(8))) float v8f;

#define K_SIZE 7
#define BATCH  64
#define T_OUT  2000
#define NCODES 1024
#define DDIM   64

#define VQ_THREADS 512          // 16 waves/block
#define VQ_WAVES   (VQ_THREADS / 32)
#define CHUNK_N    128          // codes per LDS chunk
#define LDS_ROW    33           // float2 per code row (32 + 1 pad -> 66 dwords)

// ---------------- causal strided conv1d + ELU (fp32) -----------------------
template<int CIN, int COUT, int STRIDE>
__global__ __launch_bounds__(256)
void causal_conv_elu(const float* __restrict__ x, const float* __restrict__ w,
                     const float* __restrict__ bias, float* __restrict__ y,
                     int Lin, int Lout)
{
    __shared__ float ws[CIN * K_SIZE];
    const int cout = blockIdx.y;
    const int b    = blockIdx.z;
    for (int i = threadIdx.x; i < CIN * K_SIZE; i += blockDim.x)
        ws[i] = w[(size_t)cout * CIN * K_SIZE + i];
    __syncthreads();

    const int t = blockIdx.x * blockDim.x + threadIdx.x;
    if (t >= Lout) return;

    float acc = bias[cout];
    const int base = t * STRIDE - (K_SIZE - 1);        // left-pad K-1 (causal)
    const float* xb = x + (size_t)b * CIN * Lin;
    #pragma unroll 4
    for (int ci = 0; ci < CIN; ++ci) {
        const float* xr = xb + (size_t)ci * Lin;
        const float* wr = ws + ci * K_SIZE;
        #pragma unroll
        for (int k = 0; k < K_SIZE; ++k) {
            const int idx = base + k;
            if (idx >= 0) acc = fmaf(xr[idx], wr[k], acc);
        }
    }
    // jax.nn.elu (alpha = 1)
    y[((size_t)b * COUT + cout) * Lout + t] = acc > 0.f ? acc : expm1f(acc);
}

// ---------------- per-code squared norms -----------------------------------
__global__ __launch_bounds__(256)
void code_norms(const float* __restrict__ codebooks, float* __restrict__ cnorm)
{
    const int i = blockIdx.x * blockDim.x + threadIdx.x;   // 0..2047
    const float* c = codebooks + (size_t)i * DDIM;
    float s = 0.f;
    #pragma unroll
    for (int d = 0; d < DDIM; ++d) s = fmaf(c[d], c[d], s);
    cnorm[i] = s;
}

// ---------------- VQ: WMMA f32 GEMM + fused argmin -------------------------
// One wave handles 16 points (consecutive t, one batch) x 1024 codes.
// argmin_n(||z||^2 - 2 z.c + ||c||^2) == argmin_n(||c||^2 - 2 z.c)
// 16 waves/block share LDS-staged codebook chunks (every block is single-cb).
__global__ __launch_bounds__(VQ_THREADS)
void vq_argmin(const float* __restrict__ z,          // (B,128,T) fp32
               const float* __restrict__ codebooks,  // (2,1024,64)
               const float* __restrict__ cnorm,      // (2048,)
               int* __restrict__ out)                // (2,B,T) int32
{
    __shared__ float2 ldsB[CHUNK_N * LDS_ROW];        // 33,792 bytes

    const int wave = blockIdx.x * VQ_WAVES + (threadIdx.x >> 5);
    const int lane = threadIdx.x & 31;
    const int half = lane >> 4;        // K-pair select within fragments
    const int c16  = lane & 15;        // M (for A) / N (for B,D) index

    const int cb   = wave / ((BATCH * T_OUT) / 16);      // block-uniform 0/1
    const int tile = wave - cb * ((BATCH * T_OUT) / 16);
    const int p0   = tile * 16;                           // T_OUT % 16 == 0
    const int b    = p0 / T_OUT;
    const int t0   = p0 - b * T_OUT;

    // --- A fragments: 16 K-steps of a 16x4 f32 tile (ISA 7.12.2 layout) ---
    // lane l<16:  VGPR0 = z[M=l][k0+0], VGPR1 = z[M=l][k0+1]
    // lane l>=16: VGPR0 = z[M=l-16][k0+2], VGPR1 = z[M=l-16][k0+3]
    v2f afrag[16];
    const float* zb = z + ((size_t)b * 128 + cb * DDIM) * T_OUT + t0;
    #pragma unroll
    for (int kk = 0; kk < 16; ++kk) {
        const int kA = kk * 4 + 2 * half;
        afrag[kk].x = zb[(size_t)kA * T_OUT + c16];
        afrag[kk].y = zb[(size_t)(kA + 1) * T_OUT + c16];
    }

    const float* cbase = codebooks + (size_t)cb * NCODES * DDIM;
    const float* cn    = cnorm + cb * NCODES;

    float bestS[8];
    int   bestI[8];
    #pragma unroll
    for (int i = 0; i < 8; ++i) { bestS[i] = 3.4e38f; bestI[i] = 0; }

    for (int c0 = 0; c0 < NCODES; c0 += CHUNK_N) {
        __syncthreads();   // previous chunk fully consumed
        // Cooperative load: 4096 float2 elements, coalesced 256B per 32 lanes.
        // LDS layout ldsB[n][d2] padded: stores & reads are (near) conflict-free.
        #pragma unroll
        for (int j = 0; j < (CHUNK_N * (DDIM / 2)) / VQ_THREADS; ++j) {
            const int e  = j * VQ_THREADS + threadIdx.x;
            const int n  = e >> 5;          // code within chunk
            const int d2 = e & 31;          // float2 column
            ldsB[n * LDS_ROW + d2] =
                *(const float2*)(cbase + (size_t)(c0 + n) * DDIM + 2 * d2);
        }
        __syncthreads();

        // Prefetch next chunk into cache while this chunk computes.
        if (c0 + CHUNK_N < NCODES) {
            const int pn = threadIdx.x & (CHUNK_N - 1);
            const int ps = threadIdx.x >> 7;            // 4 segments x 64B
            __builtin_prefetch(cbase + (size_t)(c0 + CHUNK_N + pn) * DDIM + ps * 16, 0, 3);
        }

        for (int nt = 0; nt < CHUNK_N; nt += 16) {
            // Load the full B tile (16 k-steps, 8x ds_load_2addr_b64) before
            // the WMMA chain. The sched_barrier stops the scheduler from
            // sinking the loads back between WMMAs (which forces full
            // s_wait_dscnt 0 drains); with all loads in flight the waiter can
            // use partial dscnt counts and overlap LDS latency with WMMA issue.
            v2f bfrag[16];
            const float2* brow = ldsB + (nt + c16) * LDS_ROW;
            #pragma unroll
            for (int kk = 0; kk < 16; ++kk) {
                const float2 bv = brow[2 * kk + half];   // C[n][4k+2h .. +1]
                bfrag[kk].x = bv.x; bfrag[kk].y = bv.y;
            }
#if __has_builtin(__builtin_amdgcn_sched_barrier)
            __builtin_amdgcn_sched_barrier(0);
#endif
            v8f acc = {};
            #pragma unroll
            for (int kk = 0; kk < 16; ++kk) {
                acc = __builtin_amdgcn_wmma_f32_16x16x4_f32(
                    /*neg_a=*/false, afrag[kk], /*neg_b=*/false, bfrag[kk],
                    /*c_mod=*/(short)0, acc, /*reuse_a=*/false, /*reuse_b=*/false);
            }
            const int   n   = c0 + nt + c16;
            const float cnv = cn[n];
            #pragma unroll
            for (int i = 0; i < 8; ++i) {
                const float s = fmaf(-2.f, acc[i], cnv);        // ||c||^2 - 2 z.c
                if (s < bestS[i]) { bestS[i] = s; bestI[i] = n; } // keeps first min
            }
        }
    }

    // min-reduce (score, index) across the 16 lanes of each half
    #pragma unroll
    for (int i = 0; i < 8; ++i) {
        float bs = bestS[i]; int bi = bestI[i];
        #pragma unroll
        for (int off = 8; off >= 1; off >>= 1) {
            const float os = __shfl_xor(bs, off, 16);
            const int   oi = __shfl_xor(bi, off, 16);
            if (os < bs || (os == bs && oi < bi)) { bs = os; bi = oi; }
        }
        bestS[i] = bs; bestI[i] = bi;
    }

    if (c16 == 0) {   // lane 0 writes M=0..7, lane 16 writes M=8..15
        int* orow = out + ((size_t)cb * BATCH + b) * T_OUT + t0 + 8 * half;
        #pragma unroll
        for (int i = 0; i < 8; ++i) orow[i] = bestI[i];
    }
}

// ---------------------------------------------------------------------------
extern "C" void kernel_launch(void* const* d_in, const int* in_sizes, int n_in,
                              void* d_out, int out_size, void* d_ws, size_t ws_size,
                              hipStream_t stream)
{
    (void)in_sizes; (void)n_in; (void)out_size; (void)ws_size;

    const float* x  = (const float*)d_in[0];
    const float* w0 = (const float*)d_in[1];  const float* b0 = (const float*)d_in[2];
    const float* w1 = (const float*)d_in[3];  const float* b1 = (const float*)d_in[4];
    const float* w2 = (const float*)d_in[5];  const float* b2 = (const float*)d_in[6];
    const float* w3 = (const float*)d_in[7];  const float* b3 = (const float*)d_in[8];
    const float* wq = (const float*)d_in[9];  const float* bq = (const float*)d_in[10];
    const float* codebooks = (const float*)d_in[11];
    int* out = (int*)d_out;

    // Workspace layout (floats): two 24.576M ping-pong buffers, z_e, cnorm.
    float* W  = (float*)d_ws;
    float* bufA = W;                        // 64*16*24000 = 24,576,000 f
    float* bufB = W + 24576000;             // same size
    float* z_e  = W + 49152000;             // 64*128*2000 = 16,384,000 f
    float* cnrm = W + 65536000;             // 2048 f   (total ~262 MB)

    const dim3 blk(256);
    causal_conv_elu<1, 16, 2><<<dim3((24000 + 255) / 256, 16, BATCH), blk, 0, stream>>>(
        x,    w0, b0, bufA, 48000, 24000);
    causal_conv_elu<16, 32, 2><<<dim3((12000 + 255) / 256, 32, BATCH), blk, 0, stream>>>(
        bufA, w1, b1, bufB, 24000, 12000);
    causal_conv_elu<32, 64, 2><<<dim3((6000 + 255) / 256, 64, BATCH), blk, 0, stream>>>(
        bufB, w2, b2, bufA, 12000, 6000);
    causal_conv_elu<64, 64, 3><<<dim3((2000 + 255) / 256, 64, BATCH), blk, 0, stream>>>(
        bufA, w3, b3, bufB, 6000, 2000);
    causal_conv_elu<64, 128, 1><<<dim3((2000 + 255) / 256, 128, BATCH), blk, 0, stream>>>(
        bufB, wq, bq, z_e, 2000, 2000);

    code_norms<<<8, 256, 0, stream>>>(codebooks, cnrm);

    // 2 codebooks * 8000 point-tiles = 16000 waves = 1000 blocks * 16 waves
    vq_argmin<<<1000, VQ_THREADS, 0, stream>>>(z_e, codebooks, cnrm, out);
}